// CenterLoss_695784702148
// MI455X (gfx1250) — compile-verified
//
#include <hip/hip_runtime.h>
#include <hip/hip_bf16.h>

#define BATCH 1024
#define NUM_CLASSES 10000
#define FEAT_DIM 512
#define KSPLIT 4
#define KCHUNK (FEAT_DIM / KSPLIT)   // 128 features per wave -> 32 wmma steps

typedef __attribute__((ext_vector_type(2))) float v2f;
typedef __attribute__((ext_vector_type(8))) float v8f;

// One wave (32 lanes) per (16-row tile, K-chunk). Computes the diagonal of
// Diff * Diff^T for Diff[16][128] via V_WMMA_F32_16X16X4_F32 accumulation.
// Lane L holds Diff[L&15][k0 + 2*(L>>4) + {0,1}] each step; this single v2f is
// simultaneously the correct A (16x4) and B (4x16) operand layout.
__global__ __launch_bounds__(32) void CenterLoss_tile_kernel(
    const float* __restrict__ x,
    const int* __restrict__ labels,
    const float* __restrict__ centers,
    float* __restrict__ part)   // part[b*KSPLIT + ky] = partial sq-dist
{
    const int tile  = blockIdx.x;          // 0..63  (16 samples each)
    const int ky    = blockIdx.y;          // 0..3   (128-feature chunk)
    const int lane  = threadIdx.x;         // 0..31
    const int m     = lane & 15;           // row within tile
    const int row   = tile * 16 + m;       // sample index
    const int khalf = (lane >> 4) << 1;    // 0 or 2

    const int lab = labels[row];
    const float* __restrict__ xp = x       + (size_t)row * FEAT_DIM + ky * KCHUNK + khalf;
    const float* __restrict__ cp = centers + (size_t)lab * FEAT_DIM + ky * KCHUNK + khalf;

    v8f acc = {};
#pragma unroll 4
    for (int i = 0; i < KCHUNK / 4; ++i) {
        v2f a;
        a.x = xp[0] - cp[0];
        a.y = xp[1] - cp[1];
        // D = A * B + C, A == B == diff chunk  (Gram matrix accumulation)
        acc = __builtin_amdgcn_wmma_f32_16x16x4_f32(
            /*neg_a=*/false, a, /*neg_b=*/false, a,
            /*c_mod=*/(short)0, acc, /*reuse_a=*/false, /*reuse_b=*/false);
        xp += 4;
        cp += 4;
    }

    // Diagonal extraction: d_m = acc[m] @ lane m (m<8); acc[m-8] @ lane m+16 (m>=8)
    const int idx = (lane < 8) ? lane : (lane - 24);
    float v = 0.0f;
#pragma unroll
    for (int i = 0; i < 8; ++i) {
        float e = acc[i];
        v = (idx == i) ? e : v;
    }

    if (lane < 8) {
        part[(size_t)(tile * 16 + lane) * KSPLIT + ky] = v;
    } else if (lane >= 24) {
        part[(size_t)(tile * 16 + (lane - 16)) * KSPLIT + ky] = v;
    }
}

// Single block: fixed-order (deterministic) reduction of 1024 samples.
// Each sample: sum 4 K-partials, clip to [1e-12, 1e12]; masked-out entries of
// the reference distmat contribute clip(0)=1e-12 each -> + (C-1)*1e-12.
__global__ __launch_bounds__(256) void CenterLoss_reduce_kernel(
    const float* __restrict__ part, float* __restrict__ out)
{
    __shared__ float s[256];
    const int t = threadIdx.x;
    float local = 0.0f;
#pragma unroll
    for (int j = 0; j < BATCH / 256; ++j) {
        const int b = t + j * 256;
        float d = ((part[b * KSPLIT + 0] + part[b * KSPLIT + 1]) +
                    part[b * KSPLIT + 2]) + part[b * KSPLIT + 3];
        d = fminf(fmaxf(d, 1e-12f), 1e12f);
        local += d;
    }
    s[t] = local;
    __syncthreads();
#pragma unroll
    for (int off = 128; off > 0; off >>= 1) {
        if (t < off) s[t] += s[t + off];
        __syncthreads();
    }
    if (t == 0) {
        out[0] = s[0] / (float)BATCH + (float)(NUM_CLASSES - 1) * 1e-12f;
    }
}

extern "C" void kernel_launch(void* const* d_in, const int* in_sizes, int n_in,
                              void* d_out, int out_size, void* d_ws, size_t ws_size,
                              hipStream_t stream) {
    const float* x       = (const float*)d_in[0];   // (1024, 512) f32
    const int*   labels  = (const int*)d_in[1];     // (1024,) int
    const float* centers = (const float*)d_in[2];   // (10000, 512) f32
    float* out  = (float*)d_out;                    // scalar loss
    float* part = (float*)d_ws;                     // 1024 * 4 floats = 16 KB

    dim3 grid(BATCH / 16, KSPLIT);   // 64 tiles x 4 K-chunks = 256 waves
    CenterLoss_tile_kernel<<<grid, 32, 0, stream>>>(x, labels, centers, part);
    CenterLoss_reduce_kernel<<<1, 256, 0, stream>>>(part, out);
}